// GATLayer_30726196036137
// MI455X (gfx1250) — compile-verified
//
#include <hip/hip_runtime.h>
#include <hip/hip_bf16.h>

// GAT layer for MI455X (gfx1250, wave32).
// Pipeline (all on `stream`):
//   1. pack_W      : W f32[256,128] -> K-pair float2 B-fragment layout (128KB, L2)
//   2. gemm_h      : h = x @ W via v_wmma_f32_16x16x4_f32  (exact f32 matrix path)
//   3. attn_logits : a_src = h . att_src ; a_dst = h . att_dst  (wave32 shfl reduce)
//   4. init_bufs   : out = bias ; segmax = enc(-inf) ; denom = 0
//   5. edge_max    : e = leakyrelu(a_src[s]+a_dst[d]); atomicMax(enc) per dst
//   6. edge_sum    : e_exp = exp(e - max[d]); atomicAdd denom[d]
//   7. edge_scat   : out[d] += (e_exp/(denom[d]+eps)) * h[s]   (f32 atomics, L2)

#define F_IN  256
#define F_OUT 128
#define NEG_SLOPE 0.2f
#define SOFTMAX_EPS 1e-16f

typedef __attribute__((ext_vector_type(2))) float v2f;
typedef __attribute__((ext_vector_type(8))) float v8f;

// Order-preserving float<->uint encoding so atomicMax(uint) == float max.
__device__ __forceinline__ unsigned int encF(float f) {
  unsigned int u = __float_as_uint(f);
  return (u & 0x80000000u) ? ~u : (u | 0x80000000u);
}
__device__ __forceinline__ float decF(unsigned int u) {
  u = (u & 0x80000000u) ? (u & 0x7FFFFFFFu) : ~u;
  return __uint_as_float(u);
}

// ---------------------------------------------------------------------------
// 1. Pack W into K-pair layout: Wp[p*128 + n] = { W[2p][n], W[2p+1][n] }.
//    The f32 WMMA B fragment for lane = (half,n) at k-step is then a single
//    contiguous 8B load at pair index (k/2 + half).
// ---------------------------------------------------------------------------
__global__ void pack_W(const float* __restrict__ W, float2* __restrict__ Wp) {
  int tid = blockIdx.x * blockDim.x + threadIdx.x;      // 128 pairs * 128 cols
  if (tid >= (F_IN / 2) * F_OUT) return;
  int n = tid & (F_OUT - 1);
  int p = tid >> 7;
  float2 v;
  v.x = W[(2 * p) * F_OUT + n];
  v.y = W[(2 * p + 1) * F_OUT + n];
  Wp[tid] = v;
}

// ---------------------------------------------------------------------------
// 2. h = x @ W.  One wave per 16 rows; 8 column tiles; K=256 in 64 steps of 4.
//    A 16x4 f32 layout: lanes 0-15 hold M=lane, K={k,k+1}; lanes 16-31 K={k+2,k+3}.
//    B 4x16: lane n holds col n with the same K-pair split.
// ---------------------------------------------------------------------------
__global__ void __launch_bounds__(256) gemm_h(const float* __restrict__ x,
                                              const float2* __restrict__ Wp,
                                              float* __restrict__ h, int N) {
  const int lane = threadIdx.x & 31;
  const int gw   = (blockIdx.x * blockDim.x + threadIdx.x) >> 5;
  const int nTiles = (N + 15) >> 4;
  if (gw >= nTiles) return;
  const int rowBase = gw << 4;
  const int half = lane >> 4;
  const int lm   = lane & 15;
  const int arow = min(rowBase + lm, N - 1);
  const float* xr = x + (size_t)arow * F_IN;

  v8f acc[8] = {};

  #pragma unroll 4
  for (int kk = 0; kk < F_IN / 4; ++kk) {
    const float2 av = *(const float2*)(xr + 4 * kk + 2 * half);
    v2f a; a[0] = av.x; a[1] = av.y;
    const float2* wb = Wp + (size_t)(2 * kk + half) * F_OUT;
    #pragma unroll
    for (int c = 0; c < 8; ++c) {
      const float2 bv = wb[c * 16 + lm];
      v2f b; b[0] = bv.x; b[1] = bv.y;
      acc[c] = __builtin_amdgcn_wmma_f32_16x16x4_f32(
          false, a, false, b, (short)0, acc[c], false, false);
    }
  }

  // D layout: lane half -> M = 8*half + j, N = lm (per ISA 16x16 f32 C/D table)
  #pragma unroll
  for (int c = 0; c < 8; ++c) {
    #pragma unroll
    for (int j = 0; j < 8; ++j) {
      int r = rowBase + half * 8 + j;
      if (r < N) h[(size_t)r * F_OUT + c * 16 + lm] = acc[c][j];
    }
  }
}

// ---------------------------------------------------------------------------
// 3. Per-node attention logits (one wave per row, shfl_xor reduce, wave32)
// ---------------------------------------------------------------------------
__global__ void __launch_bounds__(256) attn_logits(const float* __restrict__ h,
                                                   const float* __restrict__ att_src,
                                                   const float* __restrict__ att_dst,
                                                   float* __restrict__ a_src,
                                                   float* __restrict__ a_dst, int N) {
  const int lane = threadIdx.x & 31;
  const int row  = (blockIdx.x * blockDim.x + threadIdx.x) >> 5;
  if (row >= N) return;
  float s = 0.f, d = 0.f;
  #pragma unroll
  for (int c = lane; c < F_OUT; c += 32) {
    float v = h[(size_t)row * F_OUT + c];
    s = fmaf(v, att_src[c], s);
    d = fmaf(v, att_dst[c], d);
  }
  #pragma unroll
  for (int off = 16; off >= 1; off >>= 1) {
    s += __shfl_xor(s, off, 32);
    d += __shfl_xor(d, off, 32);
  }
  if (lane == 0) { a_src[row] = s; a_dst[row] = d; }
}

// ---------------------------------------------------------------------------
// 4. Init: out = bias (broadcast), segmax = enc(-inf)=0, denom = 0
// ---------------------------------------------------------------------------
__global__ void init_bufs(float* __restrict__ out, const float* __restrict__ bias,
                          unsigned int* __restrict__ segmax, float* __restrict__ denom,
                          int N) {
  int tid = blockIdx.x * blockDim.x + threadIdx.x;
  if (tid < N * F_OUT) out[tid] = bias[tid & (F_OUT - 1)];
  if (tid < N) { segmax[tid] = 0u; denom[tid] = 0.f; }
}

__device__ __forceinline__ void edge_sd(const int* __restrict__ ei, int i, int E,
                                        int* s, int* d) {
  if (i < E) { *s = ei[i]; *d = ei[E + i]; }
  else       { *s = *d = i - E; }              // self loops appended
}

// 5. e = leaky_relu(a_src[s] + a_dst[d]); segment max via encoded atomicMax
__global__ void edge_max(const int* __restrict__ ei, const float* __restrict__ a_src,
                         const float* __restrict__ a_dst, float* __restrict__ e,
                         unsigned int* __restrict__ segmax, int E, int N) {
  int i = blockIdx.x * blockDim.x + threadIdx.x;
  if (i >= E + N) return;
  int s, d; edge_sd(ei, i, E, &s, &d);
  float v = a_src[s] + a_dst[d];
  v = (v > 0.f) ? v : NEG_SLOPE * v;
  e[i] = v;
  atomicMax(&segmax[d], encF(v));
}

// 6. e_exp = exp(e - max[d]); denom[d] += e_exp
__global__ void edge_sum(const int* __restrict__ ei, float* __restrict__ e,
                         const unsigned int* __restrict__ segmax,
                         float* __restrict__ denom, int E, int N) {
  int i = blockIdx.x * blockDim.x + threadIdx.x;
  if (i >= E + N) return;
  int s, d; edge_sd(ei, i, E, &s, &d);
  float ex = __expf(e[i] - decF(segmax[d]));
  e[i] = ex;
  atomicAdd(&denom[d], ex);
}

// 7. out[d] += alpha * h[s]  (one wave per edge, float4 gather, f32 atomics)
__global__ void __launch_bounds__(256) edge_scat(const int* __restrict__ ei,
                                                 const float* __restrict__ e,
                                                 const float* __restrict__ denom,
                                                 const float* __restrict__ h,
                                                 float* __restrict__ out, int E, int N) {
  const int lane = threadIdx.x & 31;
  const int i    = (blockIdx.x * blockDim.x + threadIdx.x) >> 5;
  if (i >= E + N) return;
  int s, d; edge_sd(ei, i, E, &s, &d);
  float alpha = e[i] / (denom[d] + SOFTMAX_EPS);
  const float4 v = ((const float4*)(h + (size_t)s * F_OUT))[lane];
  float* op = out + (size_t)d * F_OUT + lane * 4;
  atomicAdd(op + 0, alpha * v.x);
  atomicAdd(op + 1, alpha * v.y);
  atomicAdd(op + 2, alpha * v.z);
  atomicAdd(op + 3, alpha * v.w);
}

// ---------------------------------------------------------------------------
extern "C" void kernel_launch(void* const* d_in, const int* in_sizes, int n_in,
                              void* d_out, int out_size, void* d_ws, size_t ws_size,
                              hipStream_t stream) {
  const float* x        = (const float*)d_in[0];
  const int*   ei       = (const int*)  d_in[1];
  const float* W        = (const float*)d_in[2];
  const float* att_src  = (const float*)d_in[3];
  const float* att_dst  = (const float*)d_in[4];
  const float* bias     = (const float*)d_in[5];
  float*       out      = (float*)d_out;

  const int N = in_sizes[0] / F_IN;    // 50000
  const int E = in_sizes[1] / 2;       // 800000
  const int T = E + N;

  // Workspace carve-up (all 16B aligned). ~30 MB total.
  char* ws = (char*)d_ws;
  size_t off = 0;
  float*        h      = (float*)(ws + off);        off += (size_t)N * F_OUT * 4;
  float2*       Wp     = (float2*)(ws + off);       off += (size_t)(F_IN / 2) * F_OUT * 8;
  float*        a_src  = (float*)(ws + off);        off += (size_t)N * 4;
  float*        a_dst  = (float*)(ws + off);        off += (size_t)N * 4;
  float*        e      = (float*)(ws + off);        off += (size_t)T * 4;
  unsigned int* segmax = (unsigned int*)(ws + off); off += (size_t)N * 4;
  float*        denom  = (float*)(ws + off);        off += (size_t)N * 4;
  (void)ws_size;

  // 1. pack W into K-pair layout
  {
    int thr = (F_IN / 2) * F_OUT;
    pack_W<<<(thr + 255) / 256, 256, 0, stream>>>(W, Wp);
  }

  // 2. GEMM: one wave (32 threads) per 16 output rows
  {
    int waves = (N + 15) / 16;
    int thr = waves * 32;
    gemm_h<<<(thr + 255) / 256, 256, 0, stream>>>(x, Wp, h, N);
  }

  // 3. logits: one wave per row
  {
    long long thr = (long long)N * 32;
    attn_logits<<<(int)((thr + 255) / 256), 256, 0, stream>>>(h, att_src, att_dst,
                                                              a_src, a_dst, N);
  }

  // 4. init out/segmax/denom
  {
    int thr = N * F_OUT;
    init_bufs<<<(thr + 255) / 256, 256, 0, stream>>>(out, bias, segmax, denom, N);
  }

  // 5-6. segment softmax
  edge_max<<<(T + 255) / 256, 256, 0, stream>>>(ei, a_src, a_dst, e, segmax, E, N);
  edge_sum<<<(T + 255) / 256, 256, 0, stream>>>(ei, e, segmax, denom, E, N);

  // 7. scatter-aggregate: one wave per edge
  {
    long long thr = (long long)T * 32;
    edge_scat<<<(int)((thr + 255) / 256), 256, 0, stream>>>(ei, e, denom, h, out, E, N);
  }
}